// Sampler3DLayer_33036888441168
// MI455X (gfx1250) — compile-verified
//
#include <hip/hip_runtime.h>
#include <hip/hip_bf16.h>

// ---------------------------------------------------------------------------
// Inverse-CDF categorical sampling, B=4 T=2048 V=32000.
//   out[row] = #{ v : rng[row] > cumsum(p[row])[v] }
// Bandwidth-bound: 1.05 GB single-pass read => ~45us floor @ 23.3 TB/s.
// One 256-thread (8-wave, wave32) block per row:
//   Phase A: NT-hinted float4 streaming, 128-elem segment sums via shfl_xor
//   Phase B: block scan of 256 segment sums, atomicMin -> crossing segment
//   Phase C: in-segment prefix scan as D = L(tri) x X via 4 chained
//            v_wmma_f32_16x16x4_f32 (scan-as-matmul), then threshold count.
//            Loads are clamped-index + cndmask (branch-free, never OOB).
// ---------------------------------------------------------------------------

typedef float v2f  __attribute__((ext_vector_type(2)));
typedef float v8f  __attribute__((ext_vector_type(8)));
typedef float f32x4 __attribute__((ext_vector_type(4)));

#define VOCAB     32000
#define NWAVE     8
#define PER_WAVE  4000      // 32000 / 8
#define FULL_IT   31        // 31 iters * 128 + 32 tail = 4000
#define SEG_LEN   128
#define TAIL_LEN  32
#define NSEG      256       // 8 waves * 32 segments

__global__ __launch_bounds__(256)
void sampler_inv_cdf_kernel(const float* __restrict__ p,
                            const float* __restrict__ rng,
                            int* __restrict__ out)
{
    __shared__ float segIncl[NSEG];
    __shared__ int   kSlot;

    const int row  = blockIdx.x;
    const int tid  = threadIdx.x;
    const int wave = tid >> 5;
    const int lane = tid & 31;

    const float* __restrict__ prow = p + (size_t)row * VOCAB;
    const float  r = rng[row];

    // ---------------- Phase A: stream + ordered 128-elem segment sums ------
    const float* __restrict__ wbase = prow + wave * PER_WAVE;

    #pragma unroll 4
    for (int i = 0; i < FULL_IT; ++i) {
        const f32x4 x = __builtin_nontemporal_load(
            (const f32x4*)(wbase + i * SEG_LEN + lane * 4));
        float s4 = (x.x + x.y) + (x.z + x.w);
        #pragma unroll
        for (int off = 16; off > 0; off >>= 1)
            s4 += __shfl_xor(s4, off, 32);
        if (lane == 0) segIncl[wave * 32 + i] = s4;
    }
    // tail segment: 32 elements (lanes 0..7 load a float4 each)
    {
        float s4 = 0.0f;
        if (lane < 8) {
            const f32x4 x = __builtin_nontemporal_load(
                (const f32x4*)(wbase + FULL_IT * SEG_LEN + lane * 4));
            s4 = (x.x + x.y) + (x.z + x.w);
        }
        #pragma unroll
        for (int off = 16; off > 0; off >>= 1)
            s4 += __shfl_xor(s4, off, 32);
        if (lane == 0) segIncl[wave * 32 + FULL_IT] = s4;
    }
    if (tid == 0) kSlot = NSEG;
    __syncthreads();

    // ---------------- Phase B: block scan of 256 segment sums --------------
    float x = segIncl[tid];
    #pragma unroll
    for (int off = 1; off < NSEG; off <<= 1) {
        float t = (tid >= off) ? segIncl[tid - off] : 0.0f;
        __syncthreads();
        x += t;
        segIncl[tid] = x;
        __syncthreads();
    }
    // first segment whose inclusive prefix reaches r
    if (x >= r) atomicMin(&kSlot, tid);
    __syncthreads();

    const int k = kSlot;
    if (k >= NSEG) {                 // rng beyond total mass (fp edge case)
        if (tid == 0) out[row] = VOCAB;
        return;
    }

    const float S          = (k > 0) ? segIncl[k - 1] : 0.0f;   // exclusive
    const int   elemsBefore = (k >> 5) * PER_WAVE + (k & 31) * SEG_LEN;
    const int   segLen      = ((k & 31) == FULL_IT) ? TAIL_LEN : SEG_LEN;
    const float* __restrict__ segBase = prow + elemsBefore;

    // ---------------- Phase C: WMMA triangular scan inside segment ---------
    if (wave == 0) {
        // X is 16x16 column-major over the segment: X[kk][j] = seg[j*16+kk].
        // D = L * X with L lower-triangular ones => D[i][j] = incl prefix of
        // column j through row i. K=16 done as 4 chained 16x16x4 f32 WMMAs.
        const int jcol   = (lane < 16) ? lane : (lane - 16);
        const int m      = jcol;                 // A-matrix row for this lane
        const int rowSel = (lane < 16) ? 0 : 2;  // K held in VGPR0 per layout

        v8f acc = {0.f, 0.f, 0.f, 0.f, 0.f, 0.f, 0.f, 0.f};
        #pragma unroll
        for (int s = 0; s < 4; ++s) {
            const int k0 = 4 * s + rowSel;       // global K for VGPR0
            const int k1 = k0 + 1;               // global K for VGPR1
            const int e0 = jcol * 16 + k0;
            const int e1 = jcol * 16 + k1;
            // Branch-free, never-OOB loads: clamp address into the segment,
            // then select 0 for padded positions (v_cndmask, no exec mask).
            const int e0c = (e0 < segLen) ? e0 : (segLen - 1);
            const int e1c = (e1 < segLen) ? e1 : (segLen - 1);
            const float x0 = segBase[e0c];
            const float x1 = segBase[e1c];
            v2f bmat;
            bmat.x = (e0 < segLen) ? x0 : 0.0f;
            bmat.y = (e1 < segLen) ? x1 : 0.0f;
            v2f amat;
            amat.x = (k0 <= m) ? 1.0f : 0.0f;    // L[m][k0]
            amat.y = (k1 <= m) ? 1.0f : 0.0f;    // L[m][k1]
            acc = __builtin_amdgcn_wmma_f32_16x16x4_f32(
                false, amat, false, bmat, (short)0, acc, false, false);
        }

        // Column totals live in D[15][j] = lane (16+j), VGPR 7.
        // Exclusive prefix of column totals for this lane's column.
        float cpe = 0.0f;
        #pragma unroll
        for (int jj = 0; jj < 16; ++jj) {
            const float tot = __shfl(acc[7], 16 + jj, 32);
            if (jj < jcol) cpe += tot;
        }

        // Count elements in this segment with S + inclusive_prefix < r.
        const int rbase = (lane < 16) ? 0 : 8;
        int cnt = 0;
        #pragma unroll
        for (int rr = 0; rr < 8; ++rr) {
            const int e = jcol * 16 + rbase + rr;
            const float csum = S + cpe + acc[rr];
            if (e < segLen && r > csum) cnt++;
        }
        #pragma unroll
        for (int off = 16; off > 0; off >>= 1)
            cnt += __shfl_xor(cnt, off, 32);

        if (lane == 0) out[row] = elemsBefore + cnt;
    }
}

extern "C" void kernel_launch(void* const* d_in, const int* in_sizes, int n_in,
                              void* d_out, int out_size, void* d_ws, size_t ws_size,
                              hipStream_t stream) {
    const float* p   = (const float*)d_in[0];   // [B, T, V] fp32
    const float* rng = (const float*)d_in[1];   // [B, T]    fp32
    int*         out = (int*)d_out;             // [B, T]    int32

    const int rows = in_sizes[1];               // B * T = 8192
    sampler_inv_cdf_kernel<<<rows, 256, 0, stream>>>(p, rng, out);
}